// SimpleTopKGate_73134703116978
// MI455X (gfx1250) — compile-verified
//
#include <hip/hip_runtime.h>
#include <hip/hip_bf16.h>
#include <stdint.h>

// ---------------------------------------------------------------------------
// SimpleTopKGate for MI455X (gfx1250, wave32, WMMA).
//   logits = x @ W^T + b + 0.1*noise            (fp32 WMMA 16x16x4, pipelined)
//   column capacity mask via exact radix-select of ranks 98303/98304
//     (level-0 16-bit histogram fused into GEMM epilogue)
//   row top-2 quantile mask + softmax
// ---------------------------------------------------------------------------

typedef __attribute__((ext_vector_type(2))) float v2f;
typedef __attribute__((ext_vector_type(4))) float v4f;
typedef __attribute__((ext_vector_type(8))) float v8f;

#define NUM_EXPERTS 64
#define IN_FEATS    1024
#define KB          64          // K-chunk staged in LDS
#define LDSPAD      4           // bank-conflict padding (stride 68 floats)

__device__ __forceinline__ uint32_t f2key(float f) {
    uint32_t u = __float_as_uint(f);
    return (u & 0x80000000u) ? ~u : (u | 0x80000000u);
}
__device__ __forceinline__ float key2f(uint32_t k) {
    uint32_t u = (k & 0x80000000u) ? (k ^ 0x80000000u) : ~k;
    return __uint_as_float(u);
}

// ---------------------------------------------------------------------------
// Kernel 1: zero histograms / init selection state
// ---------------------------------------------------------------------------
__global__ void tkg_zero_u32(uint32_t* p, int n) {
    int i = blockIdx.x * blockDim.x + threadIdx.x;
    int stride = gridDim.x * blockDim.x;
    for (; i < n; i += stride) p[i] = 0u;
}

__global__ void tkg_init_state(uint32_t* jflag, uint32_t* pref0, uint32_t* prefA,
                               uint32_t* prefB, uint32_t* rankIn,
                               uint32_t* maxA, uint32_t* minB) {
    int i = threadIdx.x;
    if (i < NUM_EXPERTS) {
        jflag[i] = 0u; pref0[i] = 0u; prefA[i] = 0u; prefB[i] = 0u;
        rankIn[i] = 0u; maxA[i] = 0u; minB[i] = 0xFFFFFFFFu;
    }
}

// ---------------------------------------------------------------------------
// Kernel 2: GEMM  logits[r][e] = sum_k x[r][k]*W[e][k] + b[e] + 0.1*noise[r][e]
//   block = 256 threads (8 waves), 128 rows per block.
//   Software-pipelined: next K-chunk is loaded to registers (NT) while the
//   current chunk is consumed by WMMA from LDS.
//   Epilogue also builds the level-0 radix-select histogram (key[31:16]).
// ---------------------------------------------------------------------------
__global__ __launch_bounds__(256) void tkg_gemm(const float* __restrict__ x,
                                                const float* __restrict__ W,
                                                const float* __restrict__ bias,
                                                const float* __restrict__ noise,
                                                float* __restrict__ logits,
                                                uint32_t* __restrict__ hist0) {
    __shared__ float xs[128 * (KB + LDSPAD)];
    __shared__ float wsh[NUM_EXPERTS * (KB + LDSPAD)];

    const int tid  = threadIdx.x;
    const int wave = tid >> 5;
    const int lane = tid & 31;
    const int l16  = lane & 15;
    const int hsel = lane >> 4;                 // 0: K {0,1}, 1: K {2,3}
    const long rowBase = (long)blockIdx.x * 128;

    v8f acc[4];
    #pragma unroll
    for (int t = 0; t < 4; ++t) acc[t] = (v8f){0.f,0.f,0.f,0.f,0.f,0.f,0.f,0.f};

    const int c4 = (tid & 15) * 4;
    const int r0 = tid >> 4;                    // 0..15

    // ---- prologue: stage chunk kb=0 into registers
    v4f xr[8], wr[4];
    #pragma unroll
    for (int i = 0; i < 8; ++i)
        xr[i] = __builtin_nontemporal_load(
            (const v4f*)(x + (rowBase + r0 + i * 16) * IN_FEATS + c4));
    #pragma unroll
    for (int i = 0; i < 4; ++i)
        wr[i] = *(const v4f*)(W + (long)(r0 + i * 16) * IN_FEATS + c4);

    for (int kb = 0; kb < IN_FEATS; kb += KB) {
        __syncthreads();                         // LDS free (prev chunk consumed)
        #pragma unroll
        for (int i = 0; i < 8; ++i)
            *(v4f*)&xs[(r0 + i * 16) * (KB + LDSPAD) + c4] = xr[i];
        #pragma unroll
        for (int i = 0; i < 4; ++i)
            *(v4f*)&wsh[(r0 + i * 16) * (KB + LDSPAD) + c4] = wr[i];
        __syncthreads();

        // issue next chunk's global loads; waited on a full chunk later
        const int kn = kb + KB;
        if (kn < IN_FEATS) {
            #pragma unroll
            for (int i = 0; i < 8; ++i)
                xr[i] = __builtin_nontemporal_load(
                    (const v4f*)(x + (rowBase + r0 + i * 16) * IN_FEATS + kn + c4));
            #pragma unroll
            for (int i = 0; i < 4; ++i)
                wr[i] = *(const v4f*)(W + (long)(r0 + i * 16) * IN_FEATS + kn + c4);
        }

        const float* xrow = &xs[(wave * 16 + l16) * (KB + LDSPAD)];
        #pragma unroll
        for (int ks = 0; ks < KB / 4; ++ks) {
            const int kk = ks * 4 + hsel * 2;
            v2f a = *(const v2f*)(xrow + kk);
            #pragma unroll
            for (int t = 0; t < 4; ++t) {
                v2f bf = *(const v2f*)(&wsh[(t * 16 + l16) * (KB + LDSPAD) + kk]);
                acc[t] = __builtin_amdgcn_wmma_f32_16x16x4_f32(
                    false, a, false, bf, (short)0, acc[t], false, false);
            }
        }
    }

    // epilogue: C/D layout -> D[m][n], m = v + 8*(lane/16), n = lane%16
    const long rowW = rowBase + wave * 16 + hsel * 8;
    #pragma unroll
    for (int t = 0; t < 4; ++t) {
        const int e  = t * 16 + l16;
        const float bv = bias[e];
        uint32_t* h = hist0 + (size_t)e * 65536;
        #pragma unroll
        for (int v = 0; v < 8; ++v) {
            const long r = rowW + v;
            float nz = __builtin_nontemporal_load(noise + r * NUM_EXPERTS + e);
            float val = acc[t][v] + bv + 0.1f * nz;
            logits[r * NUM_EXPERTS + e] = val;
            uint32_t key = f2key(val);
            atomicAdd(&h[key >> 16], 1u);        // fused level-0 histogram
        }
    }
}

// ---------------------------------------------------------------------------
// Kernel 3: scan level-0 histogram, one block per expert.
//   Finds the 16-bit bin holding ascending rank `rank` (s_lo) and decides
//   whether rank+1 (s_hi) shares the bin (joint) or diverges.
// ---------------------------------------------------------------------------
__global__ void tkg_scan0(const uint32_t* __restrict__ hist0, uint32_t rank,
                          uint32_t* jflag, uint32_t* pref0, uint32_t* prefA,
                          uint32_t* prefB, uint32_t* rankIn) {
    const int e = blockIdx.x, tid = threadIdx.x;
    const uint32_t* h = hist0 + (size_t)e * 65536;
    __shared__ uint32_t psum[256];
    __shared__ uint32_t csum[257];
    __shared__ uint32_t binS, cbS, cntS, nxtS;

    const int base = tid * 256;
    uint32_t s = 0;
    for (int i = 0; i < 256; ++i) s += h[base + i];
    psum[tid] = s;
    __syncthreads();
    if (tid == 0) {
        uint32_t c = 0;
        for (int i = 0; i < 256; ++i) { csum[i] = c; c += psum[i]; }
        csum[256] = c;
    }
    __syncthreads();
    if (csum[tid] <= rank && rank < csum[tid + 1]) {   // exactly one owner
        uint32_t c = csum[tid];
        for (int i = 0; i < 256; ++i) {
            uint32_t cnt = h[base + i];
            if (rank < c + cnt) { binS = base + i; cbS = c; cntS = cnt; break; }
            c += cnt;
        }
    }
    __syncthreads();
    const uint32_t bin0 = binS, cb = cbS, cnt = cntS;
    const bool joint = (rank + 1u) < (cb + cnt);
    if (joint) {
        if (tid == 0) { jflag[e] = 1u; pref0[e] = bin0; rankIn[e] = rank - cb; }
    } else {
        if (tid == 0) nxtS = 0xFFFFFFFFu;
        __syncthreads();
        for (int i = 0; i < 256; ++i) {
            uint32_t b = (uint32_t)(base + i);
            if (b > bin0 && h[base + i] != 0u) { atomicMin(&nxtS, b); break; }
        }
        __syncthreads();
        if (tid == 0) { jflag[e] = 0u; prefA[e] = bin0; prefB[e] = nxtS; }
    }
}

// ---------------------------------------------------------------------------
// Kernel 4: level-1 histogram (joint experts) + max/min of classes (diverged)
// ---------------------------------------------------------------------------
__global__ void tkg_hist1_minmax(const float* __restrict__ logits,
                                 const uint32_t* __restrict__ jflag,
                                 const uint32_t* __restrict__ pref0,
                                 const uint32_t* __restrict__ prefA,
                                 const uint32_t* __restrict__ prefB,
                                 uint32_t* __restrict__ hist1,
                                 uint32_t* __restrict__ maxA,
                                 uint32_t* __restrict__ minB, int n) {
    int i = blockIdx.x * blockDim.x + threadIdx.x;
    int stride = gridDim.x * blockDim.x;        // multiple of 64 -> e fixed per thread
    const int e = i & 63;
    const uint32_t jf = jflag[e], p0 = pref0[e], pA = prefA[e], pB = prefB[e];
    uint32_t* h = hist1 + (size_t)e * 65536;
    for (; i < n; i += stride) {
        __builtin_prefetch(logits + i + stride, 0, 0);   // global_prefetch_b8
        uint32_t key = f2key(logits[i]);
        uint32_t hi = key >> 16;
        if (jf) {
            if (hi == p0) atomicAdd(&h[key & 0xFFFFu], 1u);
        } else {
            if (hi == pA) atomicMax(&maxA[e], key);
            if (hi == pB) atomicMin(&minB[e], key);
        }
    }
}

// ---------------------------------------------------------------------------
// Kernel 5: scan level-1 -> exact 32-bit keys -> column threshold
//   quant[e] = f_lo + frac*(f_hi - f_lo)
// ---------------------------------------------------------------------------
__global__ void tkg_scan1(const uint32_t* __restrict__ hist1,
                          const uint32_t* __restrict__ jflag,
                          const uint32_t* __restrict__ pref0,
                          const uint32_t* __restrict__ rankIn,
                          const uint32_t* __restrict__ maxA,
                          const uint32_t* __restrict__ minB,
                          float* __restrict__ quantCol, float frac) {
    const int e = blockIdx.x, tid = threadIdx.x;
    if (jflag[e] == 0u) {                       // diverged at level 0
        if (tid == 0) {
            float fA = key2f(maxA[e]);
            float fB = key2f(minB[e]);
            quantCol[e] = fA + frac * (fB - fA);
        }
        return;
    }
    const uint32_t* h = hist1 + (size_t)e * 65536;
    const uint32_t rank = rankIn[e];
    __shared__ uint32_t psum[256];
    __shared__ uint32_t csum[257];
    __shared__ uint32_t binS, cbS, cntS, nxtS;

    const int base = tid * 256;
    uint32_t s = 0;
    for (int i = 0; i < 256; ++i) s += h[base + i];
    psum[tid] = s;
    __syncthreads();
    if (tid == 0) {
        uint32_t c = 0;
        for (int i = 0; i < 256; ++i) { csum[i] = c; c += psum[i]; }
        csum[256] = c;
    }
    __syncthreads();
    if (csum[tid] <= rank && rank < csum[tid + 1]) {
        uint32_t c = csum[tid];
        for (int i = 0; i < 256; ++i) {
            uint32_t cnt = h[base + i];
            if (rank < c + cnt) { binS = base + i; cbS = c; cntS = cnt; break; }
            c += cnt;
        }
    }
    __syncthreads();
    const uint32_t binLo = binS, cb = cbS, cnt = cntS;
    const bool joint = (rank + 1u) < (cb + cnt);
    uint32_t keyB_bin = binLo;
    if (!joint) {
        if (tid == 0) nxtS = 0xFFFFFFFFu;
        __syncthreads();
        for (int i = 0; i < 256; ++i) {
            uint32_t b = (uint32_t)(base + i);
            if (b > binLo && h[base + i] != 0u) { atomicMin(&nxtS, b); break; }
        }
        __syncthreads();
        keyB_bin = nxtS;
    }
    if (tid == 0) {
        uint32_t keyA = (pref0[e] << 16) | binLo;
        uint32_t keyB = (pref0[e] << 16) | keyB_bin;
        float fA = key2f(keyA), fB = key2f(keyB);
        quantCol[e] = fA + frac * (fB - fA);
    }
}

// ---------------------------------------------------------------------------
// Kernel 6: capacity mask -> row top-2 quantile mask -> softmax
//   one thread per token; row of 64 lives in VGPRs. Last reader of logits
//   and sole writer of out -> non-temporal.
// ---------------------------------------------------------------------------
__global__ __launch_bounds__(256) void tkg_finalize(const float* __restrict__ logits,
                                                    const float* __restrict__ quantCol,
                                                    float* __restrict__ out) {
    __shared__ float qc[NUM_EXPERTS];
    const int tid = threadIdx.x;
    if (tid < NUM_EXPERTS) qc[tid] = quantCol[tid];
    __syncthreads();

    const long token = (long)blockIdx.x * 256 + tid;
    const float* row = logits + token * NUM_EXPERTS;

    float m[NUM_EXPERTS];
    float m1 = -3.0e38f, m2 = -3.0e38f, m3 = -3.0e38f;
    #pragma unroll
    for (int i = 0; i < NUM_EXPERTS; i += 4) {
        v4f v4 = __builtin_nontemporal_load((const v4f*)(row + i));
        #pragma unroll
        for (int j = 0; j < 4; ++j) {
            float v = (v4[j] > qc[i + j]) ? v4[j] : -100000.0f;   // capacity mask
            m[i + j] = v;
            if (v > m1)      { m3 = m2; m2 = m1; m1 = v; }
            else if (v > m2) { m3 = m2; m2 = v; }
            else if (v > m3) { m3 = v; }
        }
    }
    // row quantile: q = 1 - 2/64, pos = 61.03125 -> s61 + 0.03125*(s62-s61)
    const float rq = m3 + 0.03125f * (m2 - m3);
    float vmax = -3.0e38f;
    #pragma unroll
    for (int i = 0; i < NUM_EXPERTS; ++i) {
        float v = (m[i] > rq) ? m[i] : -100000.0f;                // expert mask
        m[i] = v;
        if (v > vmax) vmax = v;
    }
    float sum = 0.f;
    #pragma unroll
    for (int i = 0; i < NUM_EXPERTS; ++i) {
        float ex = expf(m[i] - vmax);
        m[i] = ex; sum += ex;
    }
    const float inv = 1.0f / sum;
    #pragma unroll
    for (int i = 0; i < NUM_EXPERTS; i += 4) {
        v4f o;
        o[0] = m[i] * inv; o[1] = m[i + 1] * inv;
        o[2] = m[i + 2] * inv; o[3] = m[i + 3] * inv;
        __builtin_nontemporal_store(o, (v4f*)(out + token * NUM_EXPERTS + i));
    }
}

// ---------------------------------------------------------------------------
// Launch
// ---------------------------------------------------------------------------
extern "C" void kernel_launch(void* const* d_in, const int* in_sizes, int n_in,
                              void* d_out, int out_size, void* d_ws, size_t ws_size,
                              hipStream_t stream) {
    const float* x     = (const float*)d_in[0];   // [B,1024]
    const float* noise = (const float*)d_in[1];   // [B,64]
    const float* W     = (const float*)d_in[2];   // [64,1024]
    const float* bias  = (const float*)d_in[3];   // [64]
    float* out         = (float*)d_out;           // [B,64]

    const int Btok = in_sizes[0] / IN_FEATS;      // 131072
    const int n    = Btok * NUM_EXPERTS;          // 8,388,608

    // workspace layout
    float*    logits = (float*)d_ws;                                  // B*64 f32
    uint32_t* hist0  = (uint32_t*)((char*)d_ws + (size_t)n * 4);      // 64*65536
    uint32_t* hist1  = hist0 + (size_t)NUM_EXPERTS * 65536;           // 64*65536
    uint32_t* st     = hist1 + (size_t)NUM_EXPERTS * 65536;
    uint32_t* jflag  = st;
    uint32_t* pref0  = st + 64;
    uint32_t* prefA  = st + 128;
    uint32_t* prefB  = st + 192;
    uint32_t* rankIn = st + 256;
    uint32_t* maxA   = st + 320;
    uint32_t* minB   = st + 384;
    float*    quantC = (float*)(st + 448);

    // column quantile rank: k = floor(B*0.25)>=1, q = 1-k/B, pos = q*(B-1)
    int k = (int)((double)Btok * 0.25); if (k < 1) k = 1;
    double q   = 1.0 - (double)k / (double)Btok;
    double pos = q * (double)(Btok - 1);
    uint32_t rankLo = (uint32_t)pos;              // 98303
    float    frac   = (float)(pos - (double)rankLo); // 0.25

    tkg_zero_u32<<<2048, 256, 0, stream>>>(hist0, 2 * NUM_EXPERTS * 65536);
    tkg_init_state<<<1, 64, 0, stream>>>(jflag, pref0, prefA, prefB, rankIn, maxA, minB);
    tkg_gemm<<<Btok / 128, 256, 0, stream>>>(x, W, bias, noise, logits, hist0);
    tkg_scan0<<<NUM_EXPERTS, 256, 0, stream>>>(hist0, rankLo, jflag, pref0, prefA, prefB, rankIn);
    tkg_hist1_minmax<<<2048, 256, 0, stream>>>(logits, jflag, pref0, prefA, prefB,
                                               hist1, maxA, minB, n);
    tkg_scan1<<<NUM_EXPERTS, 256, 0, stream>>>(hist1, jflag, pref0, rankIn, maxA, minB,
                                               quantC, frac);
    tkg_finalize<<<Btok / 256, 256, 0, stream>>>(logits, quantC, out);
}